// MCDLoss_15461882265644
// MI455X (gfx1250) — compile-verified
//
#include <hip/hip_runtime.h>
#include <math.h>

typedef float v8f __attribute__((ext_vector_type(8)));
typedef float v2f __attribute__((ext_vector_type(2)));

#define BB 64
#define FF 128
#define TT 4096
#define KC 13
#define NN (BB * TT)          /* 262144 samples per coefficient */
#define TILES (NN / 16)       /* 16384 16-column tiles          */
#define C10   4.342944819032518f      /* 10/ln(10) */
#define LN2   0.6931471805599453f

// log1p(|v|) = log2(1+|v|) * ln2 via the hardware v_log_f32 transcendental.
// 1+|v| >= 1.0 so no denorm/negative special cases are needed; absolute error
// for tiny v is <= ~6e-8 which is negligible for the loss.
__device__ __forceinline__ float l1pabs(float v) {
    return __builtin_amdgcn_logf(1.0f + fabsf(v)) * LN2;
}

// Pack basis into LDS: ldsP[f2*32 + 2*m + j] = basis[m][2*f2+j], m<13, else 0.
// A-fragment for WMMA step k4, half h, lane-row m is the float2 at
// ldsP[(2*k4+h)*32 + 2*m]  -> bank-conflict-free across the 16 m-lanes.
__device__ __forceinline__ void fill_basis(float* ldsP) {
    for (int i = threadIdx.x; i < 2048; i += blockDim.x) {
        int f2 = i >> 5;
        int m  = (i >> 1) & 15;
        int j  = i & 1;
        int f  = 2 * f2 + j;
        float v = 0.0f;
        if (m < KC) {
            float scale = (m == 0) ? 0.08838834764831845f /* sqrt(1/128) */
                                   : 0.125f;              /* sqrt(2/128) */
            v = cosf(3.14159265358979323846f * ((float)f + 0.5f) * (float)m
                     / 128.0f) * scale;
        }
        ldsP[i] = v;
    }
}

// One 16x16 coefficient tile: C[m][t0+col] = sum_f basis[m][f]*log1p|mel[f][t]|
// via 32 x V_WMMA_F32_16X16X4_F32 (full f32 precision).
__device__ __forceinline__ v8f tile_dct(const float* __restrict__ base, int t0,
                                        const float* ldsP, int col, int half) {
    const float* p = base + t0 + col;
    v8f c = {0.f, 0.f, 0.f, 0.f, 0.f, 0.f, 0.f, 0.f};
#pragma unroll 8
    for (int k4 = 0; k4 < 32; ++k4) {
        int f = 4 * k4 + 2 * half;
        v2f b;
        b.x = l1pabs(p[(size_t)f * TT]);          // K row f   (V0)
        b.y = l1pabs(p[(size_t)(f + 1) * TT]);    // K row f+1 (V1)
        v2f a = *(const v2f*)&ldsP[(2 * k4 + half) * 32 + 2 * col];
        c = __builtin_amdgcn_wmma_f32_16x16x4_f32(false, a, false, b,
                                                  (short)0, c, false, false);
    }
    return c;
}

// ws layout (doubles): [0..15] S1 inp0, [16..31] S1 inp1,
// [32..47] S2 inp0, [48..63] S2 inp1, [64] mcd accumulator.
__global__ void k_init(double* stats) {
    if (threadIdx.x < 72) stats[threadIdx.x] = 0.0;
}

__global__ __launch_bounds__(256)
void k_stats(const float* __restrict__ mel0, const float* __restrict__ mel1,
             double* stats, float* __restrict__ cbuf, int store) {
    __shared__ float ldsP[2048];
    __shared__ float red[8][26];
    fill_basis(ldsP);
    __syncthreads();

    const int inp  = blockIdx.y;
    const float* mel = inp ? mel1 : mel0;
    float* cb = cbuf + (size_t)inp * KC * NN;

    const int lane = threadIdx.x & 31;
    const int wid  = threadIdx.x >> 5;
    const int col  = lane & 15;
    const int half = lane >> 4;
    const int wglob = blockIdx.x * (blockDim.x >> 5) + wid;
    const int nw    = gridDim.x * (blockDim.x >> 5);

    float a1[8] = {0, 0, 0, 0, 0, 0, 0, 0};
    float a2[8] = {0, 0, 0, 0, 0, 0, 0, 0};

    for (int tile = wglob; tile < TILES; tile += nw) {
        int b  = tile >> 8;            // TT/16 == 256
        int t0 = (tile & 255) << 4;
        const float* base = mel + (size_t)b * FF * TT;
        v8f c = tile_dct(base, t0, ldsP, col, half);
#pragma unroll
        for (int j = 0; j < 8; ++j) {
            float v = c[j];
            a1[j] += v;
            a2[j] += v * v;
            int m = j + 8 * half;
            if (store && m < KC)
                cb[(size_t)m * NN + (size_t)b * TT + t0 + col] = v;
        }
    }
    // Reduce across the 16 t-columns inside each half (masks stay < 16).
#pragma unroll
    for (int j = 0; j < 8; ++j) {
        for (int s = 1; s < 16; s <<= 1) {
            a1[j] += __shfl_xor(a1[j], s, 32);
            a2[j] += __shfl_xor(a2[j], s, 32);
        }
    }
    if (col == 0) {
#pragma unroll
        for (int j = 0; j < 8; ++j) {
            int m = j + 8 * half;
            if (m < KC) { red[wid][m] = a1[j]; red[wid][13 + m] = a2[j]; }
        }
    }
    __syncthreads();
    if (threadIdx.x < 26) {            // slot 0..12: S1[m], 13..25: S2[m]
        double s = 0.0;
        const int nwb = blockDim.x >> 5;
        for (int w = 0; w < nwb; ++w) s += (double)red[w][threadIdx.x];
        int slot = threadIdx.x;
        double* dst = (slot < KC) ? &stats[inp * 16 + slot]
                                  : &stats[32 + inp * 16 + (slot - KC)];
        atomicAdd(dst, s);
    }
}

__device__ __forceinline__ void load_cmvn(const double* stats,
                                          float* mean, float* istd) {
    if (threadIdx.x < 26) {
        int inp = threadIdx.x / KC, m = threadIdx.x % KC;
        double s1 = stats[inp * 16 + m];
        double s2 = stats[32 + inp * 16 + m];
        double mu = s1 / (double)NN;
        double var = (s2 - s1 * mu) / (double)(NN - 1);
        if (var < 0.0) var = 0.0;
        mean[threadIdx.x] = (float)mu;
        istd[threadIdx.x] = (float)(1.0 / (sqrt(var) + 1e-6));
    }
}

// Fast path: coefficients were spilled to ws.
__global__ __launch_bounds__(256)
void k_mcd_read(const float* __restrict__ cbuf, const double* stats,
                double* acc) {
    __shared__ float mean[26], istd[26];
    __shared__ float rsum[256];
    load_cmvn(stats, mean, istd);
    __syncthreads();

    const float* ct = cbuf;
    const float* cp = cbuf + (size_t)KC * NN;
    float lsum = 0.f;
    for (int i = blockIdx.x * blockDim.x + threadIdx.x; i < NN;
         i += gridDim.x * blockDim.x) {
        float s = 0.f;
#pragma unroll
        for (int k = 0; k < KC; ++k) {
            float dt = (ct[(size_t)k * NN + i] - mean[k]) * istd[k];
            float dp = (cp[(size_t)k * NN + i] - mean[13 + k]) * istd[13 + k];
            float d = dt - dp;
            s += d * d;
        }
        lsum += C10 * sqrtf(2.f * s);
    }
    rsum[threadIdx.x] = lsum;
    __syncthreads();
    for (int s = blockDim.x >> 1; s > 0; s >>= 1) {
        if (threadIdx.x < s) rsum[threadIdx.x] += rsum[threadIdx.x + s];
        __syncthreads();
    }
    if (threadIdx.x == 0) atomicAdd(acc, (double)rsum[0]);
}

// Fallback path: recompute both coefficient tiles from mel (shared A loads).
__global__ __launch_bounds__(256)
void k_mcd_recompute(const float* __restrict__ mel0,
                     const float* __restrict__ mel1,
                     const double* stats, double* acc) {
    __shared__ float ldsP[2048];
    __shared__ float mean[26], istd[26];
    __shared__ float wsum[8];
    fill_basis(ldsP);
    load_cmvn(stats, mean, istd);
    __syncthreads();

    const int lane = threadIdx.x & 31;
    const int wid  = threadIdx.x >> 5;
    const int col  = lane & 15;
    const int half = lane >> 4;
    const int wglob = blockIdx.x * (blockDim.x >> 5) + wid;
    const int nw    = gridDim.x * (blockDim.x >> 5);

    float lsum = 0.f;
    for (int tile = wglob; tile < TILES; tile += nw) {
        int b  = tile >> 8;
        int t0 = (tile & 255) << 4;
        const float* pt = mel0 + (size_t)b * FF * TT + t0 + col;
        const float* pp = mel1 + (size_t)b * FF * TT + t0 + col;
        v8f ctv = {0.f, 0.f, 0.f, 0.f, 0.f, 0.f, 0.f, 0.f};
        v8f cpv = {0.f, 0.f, 0.f, 0.f, 0.f, 0.f, 0.f, 0.f};
#pragma unroll 8
        for (int k4 = 0; k4 < 32; ++k4) {
            int f = 4 * k4 + 2 * half;
            v2f a = *(const v2f*)&ldsP[(2 * k4 + half) * 32 + 2 * col];
            v2f x, y;
            x.x = l1pabs(pt[(size_t)f * TT]);
            x.y = l1pabs(pt[(size_t)(f + 1) * TT]);
            y.x = l1pabs(pp[(size_t)f * TT]);
            y.y = l1pabs(pp[(size_t)(f + 1) * TT]);
            ctv = __builtin_amdgcn_wmma_f32_16x16x4_f32(false, a, false, x,
                                                        (short)0, ctv, false, false);
            cpv = __builtin_amdgcn_wmma_f32_16x16x4_f32(false, a, false, y,
                                                        (short)0, cpv, false, false);
        }
        float s = 0.f;
#pragma unroll
        for (int j = 0; j < 8; ++j) {
            int m = j + 8 * half;
            if (m < KC) {
                float dt = (ctv[j] - mean[m]) * istd[m];
                float dp = (cpv[j] - mean[13 + m]) * istd[13 + m];
                float d = dt - dp;
                s += d * d;
            }
        }
        s += __shfl_xor(s, 16, 32);        // fold the two row-halves (m 0-7, 8-15) per column
        if (half == 0) lsum += C10 * sqrtf(2.f * s);
    }
    for (int sh = 1; sh < 32; sh <<= 1) lsum += __shfl_xor(lsum, sh, 32);
    if (lane == 0) wsum[wid] = lsum;
    __syncthreads();
    if (threadIdx.x == 0) {
        float t = 0.f;
        const int nwb = blockDim.x >> 5;
        for (int w = 0; w < nwb; ++w) t += wsum[w];
        atomicAdd(acc, (double)t);
    }
}

__global__ void k_final(const double* stats, float* out) {
    out[0] = (float)(stats[64] / (double)NN);
}

extern "C" void kernel_launch(void* const* d_in, const int* in_sizes, int n_in,
                              void* d_out, int out_size, void* d_ws, size_t ws_size,
                              hipStream_t stream) {
    const float* mel_t = (const float*)d_in[0];
    const float* mel_p = (const float*)d_in[1];
    float* out = (float*)d_out;
    double* stats = (double*)d_ws;

    const size_t coff = 1024;  // cbuf starts 1 KB into ws, past the stats
    float* cbuf = (float*)((char*)d_ws + coff);
    const bool store = ws_size >= coff + (size_t)2 * KC * NN * sizeof(float);

    k_init<<<1, 128, 0, stream>>>(stats);

    dim3 g1(512, 2);                 // 4096 waves per input, 4 tiles/wave
    k_stats<<<g1, 256, 0, stream>>>(mel_t, mel_p, stats, cbuf, store ? 1 : 0);

    if (store)
        k_mcd_read<<<512, 256, 0, stream>>>(cbuf, stats, stats + 64);
    else
        k_mcd_recompute<<<512, 256, 0, stream>>>(mel_t, mel_p, stats, stats + 64);

    k_final<<<1, 1, 0, stream>>>(stats, out);
}